// freebase_node_prompt_layer_feature_weighted_sum_21534966022301
// MI455X (gfx1250) — compile-verified
//
#include <hip/hip_runtime.h>
#include <cstdint>
#include <cstddef>

// ---------------------------------------------------------------------------
// ELU(x*w) + symmetric edge scatter-add for MI455X (gfx1250, wave32).
//   emb = ELU(x * w)            (elementwise, w broadcast over rows)
//   out[dst] += emb[src]; out[src] += emb[dst]   for every edge
// emb (25.6MB) + out (25.6MB) are L2-resident (192MB L2): scatter phase is
// L2/atomic-throughput bound, not HBM bound. Wave-per-edge, float4-per-lane.
// CDNA5 paths: global_load_async_to_lds_b32 index staging (ASYNCcnt) +
// s_wait_asynccnt, global_prefetch_b8 on gather rows, readlane SGPR
// broadcast of edge indices so gathers use the saddr form.
// ---------------------------------------------------------------------------

#define D_FEAT 128
#define DV4 (D_FEAT / 4)          // 32 float4 per row == one wave32
#define THREADS 256               // 8 waves per block
#define EDGES_PER_BLOCK 256
#define EDGES_PER_WAVE 32

__device__ __forceinline__ float elu1(float v) {
  return v > 0.0f ? v : expm1f(v);
}

__device__ __forceinline__ void atom_add_f32(float* p, float v) {
  (void)__hip_atomic_fetch_add(p, v, __ATOMIC_RELAXED, __HIP_MEMORY_SCOPE_AGENT);
}

// One dword per active lane: copies the dword at global address gaddr to LDS address lds_off; tracked by ASYNCcnt.
__device__ __forceinline__ void async_ld_dword_to_lds(unsigned lds_off,
                                                      const void* gaddr) {
  asm volatile("global_load_async_to_lds_b32 %0, %1, off"
               :
               : "v"(lds_off), "v"(gaddr)
               : "memory");
}
__device__ __forceinline__ void wait_asynccnt0() {
  asm volatile("s_wait_asynccnt 0" ::: "memory");
}

// ---------------------------------------------------------------------------
// Kernel 1: emb = ELU(x * w), and zero-init out. One float4 per thread.
// ---------------------------------------------------------------------------
__global__ __launch_bounds__(THREADS) void emb_elu_zero_kernel(
    const float* __restrict__ x, const float* __restrict__ w,
    float* __restrict__ emb, float* __restrict__ out, int n_nodes) {
  int tid = blockIdx.x * blockDim.x + threadIdx.x;
  int total = n_nodes * DV4;
  if (tid >= total) return;
  int lane = tid & (DV4 - 1);  // column group 0..31

  float4 xv = reinterpret_cast<const float4*>(x)[tid];
  float4 wv = reinterpret_cast<const float4*>(w)[lane];

  float4 e;
  e.x = elu1(xv.x * wv.x);
  e.y = elu1(xv.y * wv.y);
  e.z = elu1(xv.z * wv.z);
  e.w = elu1(xv.w * wv.w);

  reinterpret_cast<float4*>(emb)[tid] = e;
  reinterpret_cast<float4*>(out)[tid] = make_float4(0.f, 0.f, 0.f, 0.f);
}

// ---------------------------------------------------------------------------
// Per-edge body: lane `lane` handles columns [4*lane, 4*lane+4).
// s, d are SGPR-uniform (readlane), so gathers take the saddr form.
// ---------------------------------------------------------------------------
__device__ __forceinline__ void edge_body(const float* __restrict__ emb,
                                          float* __restrict__ out, int s,
                                          int d, int lane) {
  float4 es = reinterpret_cast<const float4*>(emb + (size_t)s * D_FEAT)[lane];
  float4 ed = reinterpret_cast<const float4*>(emb + (size_t)d * D_FEAT)[lane];

  float* od = out + (size_t)d * D_FEAT + lane * 4;
  float* os = out + (size_t)s * D_FEAT + lane * 4;

  atom_add_f32(od + 0, es.x);
  atom_add_f32(od + 1, es.y);
  atom_add_f32(od + 2, es.z);
  atom_add_f32(od + 3, es.w);

  atom_add_f32(os + 0, ed.x);
  atom_add_f32(os + 1, ed.y);
  atom_add_f32(os + 2, ed.z);
  atom_add_f32(os + 3, ed.w);
}

// ---------------------------------------------------------------------------
// Kernel 2: edge scatter. Block stages 256 (src,dst) index pairs into LDS
// via async-to-LDS; each lane then caches its edge's indices in registers and
// the wave loop broadcasts them via readlane (no LDS traffic in hot loop).
// ---------------------------------------------------------------------------
__global__ __launch_bounds__(THREADS) void edge_scatter_kernel(
    const float* __restrict__ emb, const int* __restrict__ src,
    const int* __restrict__ dst, float* __restrict__ out, int n_edges) {
  __shared__ int lds_idx[2 * EDGES_PER_BLOCK];  // [0..255]=src, [256..511]=dst

  const int t = threadIdx.x;
  const int base = blockIdx.x * EDGES_PER_BLOCK;
  const int e0 = base + t;

  // ---- stage indices into LDS (CDNA5 async DMA path, ASYNCcnt) ----
  if (e0 < n_edges) {
    async_ld_dword_to_lds((unsigned)(uintptr_t)&lds_idx[t], src + e0);
    async_ld_dword_to_lds((unsigned)(uintptr_t)&lds_idx[EDGES_PER_BLOCK + t],
                          dst + e0);
  }
  wait_asynccnt0();
  __syncthreads();

  const int wave = t >> 5;
  const int lane = t & 31;
  const int wbase = wave * EDGES_PER_WAVE;

  // Lane l caches the indices of edge (base + wbase + l); garbage beyond the
  // tail is never dereferenced (guarded below).
  int sreg = lds_idx[wbase + lane];
  int dreg = lds_idx[EDGES_PER_BLOCK + wbase + lane];

  // ---- prefetch first cacheline of every gather row this wave will touch ----
  if (base + wbase + lane < n_edges) {
    __builtin_prefetch(emb + (size_t)sreg * D_FEAT, 0, 0);
    __builtin_prefetch(emb + (size_t)dreg * D_FEAT, 0, 0);
  }

  if (base + EDGES_PER_BLOCK <= n_edges) {
    // Fast path: full tile, no per-edge guards; unroll for MLP across the
    // L2 gather latency (atomics are fire-and-forget on STOREcnt).
#pragma unroll 4
    for (int i = 0; i < EDGES_PER_WAVE; ++i) {
      int s = __builtin_amdgcn_readlane(sreg, i);
      int d = __builtin_amdgcn_readlane(dreg, i);
      edge_body(emb, out, s, d, lane);
    }
  } else {
    for (int i = 0; i < EDGES_PER_WAVE; ++i) {
      if (base + wbase + i >= n_edges) break;
      int s = __builtin_amdgcn_readlane(sreg, i);
      int d = __builtin_amdgcn_readlane(dreg, i);
      edge_body(emb, out, s, d, lane);
    }
  }
}

// ---------------------------------------------------------------------------
// Launcher. Inputs: x (n*128 f32), weight (128 f32), src (E i32), dst (E i32).
// d_out: n*128 f32. d_ws: emb scratch (n*128 f32 = 25.6 MB).
// ---------------------------------------------------------------------------
extern "C" void kernel_launch(void* const* d_in, const int* in_sizes, int n_in,
                              void* d_out, int out_size, void* d_ws,
                              size_t ws_size, hipStream_t stream) {
  (void)n_in;
  (void)out_size;
  (void)ws_size;

  const float* x = (const float*)d_in[0];
  const float* w = (const float*)d_in[1];
  const int* src = (const int*)d_in[2];
  const int* dst = (const int*)d_in[3];
  float* out = (float*)d_out;
  float* emb = (float*)d_ws;

  const int n_nodes = in_sizes[0] / D_FEAT;
  const int n_edges = in_sizes[2];

  const int total_v = n_nodes * DV4;
  const int blocks1 = (total_v + THREADS - 1) / THREADS;
  emb_elu_zero_kernel<<<blocks1, THREADS, 0, stream>>>(x, w, emb, out,
                                                       n_nodes);

  const int blocks2 = (n_edges + EDGES_PER_BLOCK - 1) / EDGES_PER_BLOCK;
  edge_scatter_kernel<<<blocks2, THREADS, 0, stream>>>(emb, src, dst, out,
                                                       n_edges);
}